// EnhancedAutoformer_54812372632165
// MI455X (gfx1250) — compile-verified
//
#include <hip/hip_runtime.h>
#include <hip/hip_bf16.h>

// Problem constants (fixed by the reference)
#define B_  64
#define L_  2048
#define D_  512
#define H_  256
#define MAXK 50
#define P_  25

typedef float v2f __attribute__((ext_vector_type(2)));
typedef float v8f __attribute__((ext_vector_type(8)));

// ---------------------------------------------------------------------------
// Kernel A: x_global[b,d] = mean over L of x[b,l,d]
// ---------------------------------------------------------------------------
__global__ __launch_bounds__(256) void k_mean(const float* __restrict__ x,
                                              float* __restrict__ xg) {
    int t = blockIdx.x * 256 + threadIdx.x;           // 0 .. B*D-1
    int b = t >> 9;                                    // /512
    int d = t & (D_ - 1);
    const float* p = x + ((size_t)b * L_) * D_ + d;
    float s = 0.f;
    #pragma unroll 8
    for (int l = 0; l < L_; ++l) s += p[(size_t)l * D_];
    xg[t] = s * (1.0f / (float)L_);
}

// ---------------------------------------------------------------------------
// Kernel B: h = relu(xg @ w1 + b1)  (64x512x256) via V_WMMA_F32_16X16X4_F32
// one wave per 16x16 output tile; grid = (64/16)*(256/16) = 64 waves.
// ---------------------------------------------------------------------------
__global__ __launch_bounds__(32) void k_gemm_wmma(const float* __restrict__ xg,
                                                  const float* __restrict__ w1,
                                                  const float* __restrict__ b1,
                                                  float* __restrict__ h) {
    const int tile = blockIdx.x;
    const int mt = tile >> 4;        // 0..3
    const int nt = tile & 15;        // 0..15
    const int lane = threadIdx.x;
    const int half = lane >> 4;      // 0/1
    const int lm = lane & 15;

    const int row = mt * 16 + lm;    // M index for A frag
    const int col = nt * 16 + lm;    // N index for B frag

    v8f acc = {};
    for (int k0 = 0; k0 < D_; k0 += 4) {
        const int ka = k0 + 2 * half;
        v2f a, bfrag;
        a.x = xg[row * D_ + ka + 0];
        a.y = xg[row * D_ + ka + 1];
        bfrag.x = w1[(ka + 0) * H_ + col];
        bfrag.y = w1[(ka + 1) * H_ + col];
        acc = __builtin_amdgcn_wmma_f32_16x16x4_f32(
            false, a, false, bfrag, (short)0, acc, false, false);
    }
    #pragma unroll
    for (int r = 0; r < 8; ++r) {
        int M = mt * 16 + r + 8 * half;
        int N = col;
        float v = acc[r] + b1[N];
        h[M * H_ + N] = v > 0.f ? v : 0.f;
    }
}

// ---------------------------------------------------------------------------
// Kernel C: sig = sigmoid(h @ w2 + b2); k[b] = odd-clipped round(sig*45+5)
// ---------------------------------------------------------------------------
__global__ __launch_bounds__(64) void k_ksel(const float* __restrict__ h,
                                             const float* __restrict__ w2,
                                             const float* __restrict__ b2,
                                             int* __restrict__ kb) {
    int b = threadIdx.x;
    if (b >= B_) return;
    float z = b2[0];
    #pragma unroll 8
    for (int j = 0; j < H_; ++j) z += h[b * H_ + j] * w2[j];
    float sig = 1.0f / (1.0f + expf(-z));
    float kf = sig * (float)(MAXK - 5) + 5.0f;
    int kk = (int)rintf(kf);                 // round half-to-even (RNE)
    kk = min(max(kk, 3), MAXK);              // clip(3, min(maxK, L/2)=50)
    if ((kk & 1) == 0) kk -= 1;              // force odd
    if (kk < 3) kk = 3;
    kb[b] = kk;
}

// ---------------------------------------------------------------------------
// Kernel D: build centered, softmax-normalized kernels
// Wfull[b,d,mm] = softmax(tw[d,:k[b]])[mm - (P - k/2)] (0 outside)
// ---------------------------------------------------------------------------
__global__ __launch_bounds__(256) void k_wbuild(const float* __restrict__ tw,
                                                const int* __restrict__ kb,
                                                float* __restrict__ Wf) {
    int t = blockIdx.x * 256 + threadIdx.x;   // 0 .. B*D-1
    int b = t >> 9;
    int d = t & (D_ - 1);
    int k = kb[b];
    const float* row = tw + d * MAXK;
    float mx = -3.4e38f;
    for (int j = 0; j < k; ++j) mx = fmaxf(mx, row[j]);
    float sum = 0.f;
    for (int j = 0; j < k; ++j) sum += expf(row[j] - mx);
    float inv = 1.0f / sum;
    int off = P_ - (k >> 1);                  // center offset (k odd)
    float* out = Wf + ((size_t)b * D_ + d) * MAXK;
    #pragma unroll
    for (int mm = 0; mm < MAXK; ++mm) out[mm] = 0.f;
    for (int j = 0; j < k; ++j) out[j + off] = expf(row[j] - mx) * inv;
}

// ---------------------------------------------------------------------------
// Kernel E: depthwise variable-kernel moving average + residual
//   trend[b,l,d] = sum_m x[b, clamp(l+m-P,0,L-1), d] * Wf[b,d,m]
//   seasonal = x - trend
// Block: 256 thr = 64 d-lanes x 4 l-groups.
// Double-buffered LDS tiles (2 x 177 x 64 f32 = 90.6 KB) filled with
// GLOBAL_LOAD_ASYNC_TO_LDS_B128 (ASYNCcnt) so the next chunk's 45 KB fetch
// overlaps the current chunk's ~1800 VALU ops/thread. Taps live in VGPRs;
// each 8-accumulator pass does 57 ds_loads feeding 400 v_fmac.
// ---------------------------------------------------------------------------
#define DT 64
#define LBLK 128
#define HALO (MAXK - 1)           // 49
#define ROWS (LBLK + HALO)        // 177
#define NCHUNK (L_ / LBLK)        // 16

// Issue one tile fill as per-lane async B128 copies (16B per op).
__device__ __forceinline__ void async_fill_tile(const float* __restrict__ xb,
                                                int l0, int d0, int tid,
                                                unsigned lds_base_bytes) {
    for (int i = tid; i < ROWS * 16; i += 256) {
        int r   = i >> 4;                 // tile row
        int seg = (i & 15) << 2;          // float offset within 64-wide row
        int gl  = l0 + r - P_;            // replicate padding
        gl = min(max(gl, 0), L_ - 1);
        const float* gp = xb + (size_t)gl * D_ + d0 + seg;
        unsigned la = lds_base_bytes + (unsigned)((r * DT + seg) * 4);
        asm volatile("global_load_async_to_lds_b128 %0, %1, off"
                     :: "v"(la), "v"(gp)
                     : "memory");
    }
}

__device__ __forceinline__ void wait_async_all() {
    asm volatile("s_wait_asynccnt 0x0" ::: "memory");
}

__global__ __launch_bounds__(256) void k_conv(const float* __restrict__ x,
                                              const float* __restrict__ Wf,
                                              float* __restrict__ seasonal,
                                              float* __restrict__ trend) {
    __shared__ float xs[2][ROWS * DT];        // 2 x 45.3 KB
    const int b  = blockIdx.y;
    const int d0 = blockIdx.x * DT;
    const int tid = threadIdx.x;
    const int dd = tid & (DT - 1);
    const int g  = tid >> 6;                  // l-group 0..3
    const int d  = d0 + dd;

    const unsigned base0 = (unsigned)(uintptr_t)(&xs[0][0]);
    const unsigned base1 = (unsigned)(uintptr_t)(&xs[1][0]);

    // taps -> registers (fully unrolled => static VGPR indexing)
    float wt[MAXK];
    const float* wp = Wf + ((size_t)b * D_ + d) * MAXK;
    #pragma unroll
    for (int m = 0; m < MAXK; ++m) wt[m] = wp[m];

    const float* xb = x + (size_t)b * L_ * D_;
    float* sb = seasonal + (size_t)b * L_ * D_;
    float* tb = trend    + (size_t)b * L_ * D_;

    // prologue: fill tile 0
    async_fill_tile(xb, 0, d0, tid, base0);
    wait_async_all();
    __syncthreads();

    int cur = 0;
    for (int c = 0; c < NCHUNK; ++c) {
        const int l0 = c * LBLK;
        // prefetch next tile into the other buffer (safe: that buffer's
        // readers all passed the previous barrier)
        if (c + 1 < NCHUNK)
            async_fill_tile(xb, l0 + LBLK, d0, tid, cur ? base0 : base1);

        const float* buf = xs[cur];

        #pragma unroll 1
        for (int s = 0; s < 4; ++s) {
            const int lo = g * 32 + s * 8;    // first output row of this pass
            float acc[8];
            #pragma unroll
            for (int a = 0; a < 8; ++a) acc[a] = 0.f;
            #pragma unroll
            for (int t = 0; t < 8 + HALO; ++t) {      // 57 sliding loads
                float xv = buf[(lo + t) * DT + dd];
                #pragma unroll
                for (int a = 0; a < 8; ++a) {
                    int m = t - a;
                    if (m >= 0 && m < MAXK)
                        acc[a] = fmaf(xv, wt[m], acc[a]);
                }
            }
            #pragma unroll
            for (int a = 0; a < 8; ++a) {
                int gl = l0 + lo + a;
                float xv = buf[(lo + a + P_) * DT + dd];   // x[b,gl,d]
                size_t o = (size_t)gl * D_ + d;
                tb[o] = acc[a];
                sb[o] = xv - acc[a];
            }
        }

        wait_async_all();      // my next-tile fills have landed
        __syncthreads();       // everyone done reading cur + filling cur^1
        cur ^= 1;
    }
}

// ---------------------------------------------------------------------------
extern "C" void kernel_launch(void* const* d_in, const int* in_sizes, int n_in,
                              void* d_out, int out_size, void* d_ws, size_t ws_size,
                              hipStream_t stream) {
    const float* x  = (const float*)d_in[0];   // [B,L,D]
    const float* tw = (const float*)d_in[1];   // [D,MAXK]
    const float* w1 = (const float*)d_in[2];   // [D,H]
    const float* b1 = (const float*)d_in[3];   // [H]
    const float* w2 = (const float*)d_in[4];   // [H,1]
    const float* b2 = (const float*)d_in[5];   // [1]

    float* seasonal = (float*)d_out;
    float* trend    = seasonal + (size_t)B_ * L_ * D_;

    float* ws = (float*)d_ws;
    float* xg = ws;                                   // B*D      = 32768
    float* h  = ws + (size_t)B_ * D_;                 // B*H      = 16384
    float* Wf = h  + (size_t)B_ * H_;                 // B*D*MAXK = 1638400
    int*   kb = (int*)(Wf + (size_t)B_ * D_ * MAXK);  // B ints

    k_mean<<<(B_ * D_) / 256, 256, 0, stream>>>(x, xg);
    k_gemm_wmma<<<(B_ / 16) * (H_ / 16), 32, 0, stream>>>(xg, w1, b1, h);
    k_ksel<<<1, 64, 0, stream>>>(h, w2, b2, kb);
    k_wbuild<<<(B_ * D_) / 256, 256, 0, stream>>>(tw, kb, Wf);
    dim3 cgrid(D_ / DT, B_);
    k_conv<<<cgrid, 256, 0, stream>>>(x, Wf, seasonal, trend);
}